// GAT_FULL_22316650070209
// MI455X (gfx1250) — compile-verified
//
#include <hip/hip_runtime.h>
#include <hip/hip_bf16.h>
#include <math.h>

#define N_NODES 50000
#define N_EDGES 800000
#define IN_SIZE 128
#define HID 32
#define NHEAD1 8
#define OUT_D 40
#define NHEAD2 1
#define NEG_SLOPE 0.2f

typedef float v2f __attribute__((ext_vector_type(2)));
typedef float v8f __attribute__((ext_vector_type(8)));

// ---- order-preserving float<->int mapping for atomicMax on f32 ----
static __device__ __forceinline__ int ord_of_float(float f) {
  int i = __float_as_int(f);
  return (i >= 0) ? i : (i ^ 0x7fffffff);
}
static __device__ __forceinline__ float float_of_ord(int o) {
  int i = (o >= 0) ? o : (o ^ 0x7fffffff);
  return __int_as_float(i);
}
#define ORD_NEG_INF 0x807fffffu /* == ord_of_float(-inf) */

// ---------------- generic fill ----------------
__global__ void fill_u32_kernel(unsigned* __restrict__ p, unsigned v, long long n) {
  long long i = (long long)blockIdx.x * blockDim.x + threadIdx.x;
  if (i < n) p[i] = v;
}

// ---------------- zero-pad copy: dst[rows,dn] <- src[rows,sn] ----------------
__global__ void pad_copy_kernel(const float* __restrict__ src, float* __restrict__ dst,
                                int rows, int sn, int dn) {
  int idx = blockIdx.x * blockDim.x + threadIdx.x;
  if (idx >= rows * dn) return;
  int r = idx / dn, c = idx - r * dn;
  dst[idx] = (c < sn) ? src[r * sn + c] : 0.0f;
}

// ---------------- WMMA fp32 GEMM: C[M,N] = A[M,K] * B[K,NB(>=N)] ----------------
// One wave per 16x16 tile. K, NB compile-time (immediate-offset loads, no
// divergence in the K-loop; EXEC all-ones for every WMMA). M multiple of 16.
// B must be padded to NB columns (NB multiple of 16). Store masked to true N.
template <int K, int NB>
__global__ void gemm_wmma_f32_kernel(const float* __restrict__ A,
                                     const float* __restrict__ B,
                                     float* __restrict__ C, int N) {
  const int lane = threadIdx.x & 31;
  const int g = lane >> 4;  // lane group: 0 -> K {k,k+1}, 1 -> K {k+2,k+3}
  const int l = lane & 15;
  const int m0 = blockIdx.x * 16;
  const int n0 = blockIdx.y * 16;
  const int n = n0 + l;

  const float* a = A + (size_t)(m0 + l) * K + 2 * g;
  const float* b = B + (size_t)(2 * g) * NB + n;  // padded: always in-bounds

  v8f acc = {};
#pragma unroll 8
  for (int k = 0; k < K; k += 4) {
    v2f av, bv;
    av.x = a[k];
    av.y = a[k + 1];
    bv.x = b[(size_t)k * NB];
    bv.y = b[(size_t)(k + 1) * NB];
    acc = __builtin_amdgcn_wmma_f32_16x16x4_f32(false, av, false, bv,
                                                (short)0, acc, false, false);
  }
  // C/D layout: VGPR r, lane group g -> row M = m0 + r + 8*g, col N = n
  if (n < N) {
    float* c = C + (size_t)(m0 + 8 * g) * N + n;
#pragma unroll
    for (int r = 0; r < 8; ++r) c[(size_t)r * N] = acc[r];
  }
}

// ---------------- el/er = sum_d z[n,h,d] * a{l,r}[h,d] ----------------
__global__ void attn_proj_kernel(const float* __restrict__ z,
                                 const float* __restrict__ al,
                                 const float* __restrict__ ar,
                                 float* __restrict__ el, float* __restrict__ er,
                                 int H, int D) {
  int idx = blockIdx.x * blockDim.x + threadIdx.x;
  if (idx >= N_NODES * H) return;
  int h = idx % H;
  const float* zp = z + (size_t)idx * (size_t)D;  // idx = n*H + h, stride D
  const float* alp = al + h * D;
  const float* arp = ar + h * D;
  float sl = 0.f, sr = 0.f;
  for (int d = 0; d < D; ++d) {
    float v = zp[d];
    sl += v * alp[d];
    sr += v * arp[d];
  }
  el[idx] = sl;
  er[idx] = sr;
}

// ---------------- e = leaky_relu(el[src]+er[dst]); segment max ----------------
__global__ void edge_score_kernel(const int* __restrict__ src,
                                  const int* __restrict__ dst,
                                  const float* __restrict__ el,
                                  const float* __restrict__ er,
                                  float* __restrict__ ebuf,
                                  int* __restrict__ mmax, int H) {
  int idx = blockIdx.x * blockDim.x + threadIdx.x;
  if (idx >= N_EDGES * H) return;
  int e = idx / H, h = idx - e * H;
  int s = src[e], d = dst[e];
  float x = el[s * H + h] + er[d * H + h];
  x = (x > 0.f) ? x : NEG_SLOPE * x;
  ebuf[idx] = x;
  atomicMax(&mmax[d * H + h], ord_of_float(x));
}

// unmap ord ints to floats in place; non-finite (no in-edges) -> 0
__global__ void finalize_max_kernel(int* __restrict__ m, int count) {
  int i = blockIdx.x * blockDim.x + threadIdx.x;
  if (i >= count) return;
  float f = float_of_ord(m[i]);
  if (!isfinite(f)) f = 0.0f;
  ((float*)m)[i] = f;
}

// ---------------- ex = exp(e - m[dst]); segment sum ----------------
__global__ void edge_exp_kernel(const int* __restrict__ dst,
                                const float* __restrict__ m,
                                float* __restrict__ ebuf,
                                float* __restrict__ denom, int H) {
  int idx = blockIdx.x * blockDim.x + threadIdx.x;
  if (idx >= N_EDGES * H) return;
  int e = idx / H, h = idx - e * H;
  int d = dst[e];
  float ex = expf(ebuf[idx] - m[d * H + h]);
  ebuf[idx] = ex;
  atomicAdd(&denom[d * H + h], ex);
}

// -------- agg[dst] += (ex/denom[dst]) * z[src], float4 per thread --------
__global__ void edge_agg4_kernel(const int* __restrict__ src,
                                 const int* __restrict__ dst,
                                 const float* __restrict__ ebuf,
                                 const float* __restrict__ denom,
                                 const float* __restrict__ z,
                                 float* __restrict__ agg, int H, int D) {
  const int HD = H * D;
  const int HD4 = HD >> 2;
  long long idx = (long long)blockIdx.x * blockDim.x + threadIdx.x;
  long long total = (long long)N_EDGES * HD4;
  if (idx >= total) return;
  int e = (int)(idx / HD4);
  int q = (int)(idx - (long long)e * HD4);  // which float4 chunk within H*D
  int h = (q * 4) / D;
  int s = src[e], d = dst[e];
  float alpha = ebuf[e * H + h] / denom[d * H + h];
  float4 zv = ((const float4*)(z + (size_t)s * HD))[q];
  float* ap = agg + (size_t)d * HD + (size_t)q * 4;
  atomicAdd(ap + 0, alpha * zv.x);
  atomicAdd(ap + 1, alpha * zv.y);
  atomicAdd(ap + 2, alpha * zv.z);
  atomicAdd(ap + 3, alpha * zv.w);
}

// ---------------- h = elu(agg + b), in place ----------------
__global__ void elu_bias_kernel(float* __restrict__ x, const float* __restrict__ b, int F) {
  long long idx = (long long)blockIdx.x * blockDim.x + threadIdx.x;
  long long total = (long long)N_NODES * F;
  if (idx >= total) return;
  float v = x[idx] + b[idx % F];
  x[idx] = (v > 0.f) ? v : expm1f(v);
}

// ---------------- out += b (mean over H2=1 heads is identity) ----------------
__global__ void bias_add_kernel(float* __restrict__ out, const float* __restrict__ b, int F) {
  long long idx = (long long)blockIdx.x * blockDim.x + threadIdx.x;
  long long total = (long long)N_NODES * F;
  if (idx >= total) return;
  out[idx] += b[idx % F];
}

static inline int cdiv(long long a, int b) { return (int)((a + b - 1) / b); }

extern "C" void kernel_launch(void* const* d_in, const int* in_sizes, int n_in,
                              void* d_out, int out_size, void* d_ws, size_t ws_size,
                              hipStream_t stream) {
  const float* feat = (const float*)d_in[0];
  const int*   src  = (const int*)d_in[1];
  const int*   dst  = (const int*)d_in[2];
  const float* W1   = (const float*)d_in[3];
  const float* al1  = (const float*)d_in[4];
  const float* ar1  = (const float*)d_in[5];
  const float* b1   = (const float*)d_in[6];
  const float* W2   = (const float*)d_in[7];
  const float* al2  = (const float*)d_in[8];
  const float* ar2  = (const float*)d_in[9];
  const float* b2   = (const float*)d_in[10];
  float* out = (float*)d_out;

  // ---- workspace layout ----
  char* ws = (char*)d_ws;
  size_t off = 0;
  auto alloc = [&](size_t bytes) -> char* {
    char* p = ws + off;
    off += (bytes + 255) & ~(size_t)255;
    return p;
  };
  const int F1 = NHEAD1 * HID;  // 256
  const int OUT_P = 48;         // padded W2 column count (multiple of 16)
  float* z1   = (float*)alloc((size_t)N_NODES * F1 * 4);
  float* agg1 = (float*)alloc((size_t)N_NODES * F1 * 4);
  float* el1  = (float*)alloc((size_t)N_NODES * NHEAD1 * 4);
  float* er1  = (float*)alloc((size_t)N_NODES * NHEAD1 * 4);
  int*   m1   = (int*)  alloc((size_t)N_NODES * NHEAD1 * 4);
  float* den1 = (float*)alloc((size_t)N_NODES * NHEAD1 * 4);
  float* e1   = (float*)alloc((size_t)N_EDGES * NHEAD1 * 4);
  float* z2   = (float*)alloc((size_t)N_NODES * OUT_D * 4);
  float* W2p  = (float*)alloc((size_t)F1 * OUT_P * 4);
  float* el2  = (float*)alloc((size_t)N_NODES * 4);
  float* er2  = (float*)alloc((size_t)N_NODES * 4);
  int*   m2   = (int*)  alloc((size_t)N_NODES * 4);
  float* den2 = (float*)alloc((size_t)N_NODES * 4);
  float* e2   = (float*)alloc((size_t)N_EDGES * 4);
  (void)ws_size; (void)in_sizes; (void)n_in; (void)out_size;

  const int T = 256;

  // ================= Layer 1 =================
  fill_u32_kernel<<<cdiv((long long)N_NODES * F1, T), T, 0, stream>>>(
      (unsigned*)agg1, 0u, (long long)N_NODES * F1);
  fill_u32_kernel<<<cdiv((long long)N_NODES * NHEAD1, T), T, 0, stream>>>(
      (unsigned*)m1, ORD_NEG_INF, (long long)N_NODES * NHEAD1);
  fill_u32_kernel<<<cdiv((long long)N_NODES * NHEAD1, T), T, 0, stream>>>(
      (unsigned*)den1, 0u, (long long)N_NODES * NHEAD1);

  // z1 = feat @ W1   [50000,128]x[128,256]
  {
    dim3 grid(N_NODES / 16, F1 / 16);
    gemm_wmma_f32_kernel<IN_SIZE, 256><<<grid, 32, 0, stream>>>(feat, W1, z1, F1);
  }
  attn_proj_kernel<<<cdiv((long long)N_NODES * NHEAD1, T), T, 0, stream>>>(
      z1, al1, ar1, el1, er1, NHEAD1, HID);
  edge_score_kernel<<<cdiv((long long)N_EDGES * NHEAD1, T), T, 0, stream>>>(
      src, dst, el1, er1, e1, m1, NHEAD1);
  finalize_max_kernel<<<cdiv((long long)N_NODES * NHEAD1, T), T, 0, stream>>>(
      m1, N_NODES * NHEAD1);
  edge_exp_kernel<<<cdiv((long long)N_EDGES * NHEAD1, T), T, 0, stream>>>(
      dst, (const float*)m1, e1, den1, NHEAD1);
  edge_agg4_kernel<<<cdiv((long long)N_EDGES * NHEAD1 * HID / 4, T), T, 0, stream>>>(
      src, dst, e1, den1, z1, agg1, NHEAD1, HID);
  // h1 = elu(agg1 + b1) in place
  elu_bias_kernel<<<cdiv((long long)N_NODES * F1, T), T, 0, stream>>>(agg1, b1, F1);

  // ================= Layer 2 =================
  fill_u32_kernel<<<cdiv((long long)N_NODES * OUT_D, T), T, 0, stream>>>(
      (unsigned*)out, 0u, (long long)N_NODES * OUT_D);
  fill_u32_kernel<<<cdiv((long long)N_NODES, T), T, 0, stream>>>(
      (unsigned*)m2, ORD_NEG_INF, (long long)N_NODES);
  fill_u32_kernel<<<cdiv((long long)N_NODES, T), T, 0, stream>>>(
      (unsigned*)den2, 0u, (long long)N_NODES);
  // W2p[256,48] = zero-padded W2[256,40]
  pad_copy_kernel<<<cdiv((long long)F1 * OUT_P, T), T, 0, stream>>>(
      W2, W2p, F1, OUT_D, OUT_P);

  // z2 = h1 @ W2    [50000,256]x[256,40] (B padded to 48)
  {
    dim3 grid(N_NODES / 16, OUT_P / 16);
    gemm_wmma_f32_kernel<256, 48><<<grid, 32, 0, stream>>>(agg1, W2p, z2, OUT_D);
  }
  attn_proj_kernel<<<cdiv((long long)N_NODES * NHEAD2, T), T, 0, stream>>>(
      z2, al2, ar2, el2, er2, NHEAD2, OUT_D);
  edge_score_kernel<<<cdiv((long long)N_EDGES * NHEAD2, T), T, 0, stream>>>(
      src, dst, el2, er2, e2, m2, NHEAD2);
  finalize_max_kernel<<<cdiv((long long)N_NODES, T), T, 0, stream>>>(m2, N_NODES);
  edge_exp_kernel<<<cdiv((long long)N_EDGES * NHEAD2, T), T, 0, stream>>>(
      dst, (const float*)m2, e2, den2, NHEAD2);
  edge_agg4_kernel<<<cdiv((long long)N_EDGES * NHEAD2 * OUT_D / 4, T), T, 0, stream>>>(
      src, dst, e2, den2, z2, out, NHEAD2, OUT_D);
  // out += b2 ; mean over H2=1 heads is identity
  bias_add_kernel<<<cdiv((long long)N_NODES * OUT_D, T), T, 0, stream>>>(out, b2, OUT_D);
}